// AndersonAcceleration_9663676416061
// MI455X (gfx1250) — compile-verified
//
#include <hip/hip_runtime.h>
#include <math.h>

#define B_ 8
#define C_ 3
#define H_ 384
#define W_ 384
#define D_ (C_*H_*W_)               // 442368 elements per batch
#define BD_ ((size_t)B_*(size_t)D_) // 3538944 elements per "field"
#define HIST_ 5
#define MAXIT_ 12
#define RIDGE_ 1e-4f
#define TAU_ 0.5f

typedef __attribute__((ext_vector_type(2))) float v2f;
typedef __attribute__((ext_vector_type(4))) float v4f;
typedef __attribute__((ext_vector_type(8))) float v8f;

// ---------------- elementwise helpers ----------------

__global__ void copy4_kernel(const float* __restrict__ src, float* __restrict__ dst, size_t n4) {
  size_t t = (size_t)blockIdx.x * blockDim.x + threadIdx.x;
  if (t < n4) ((v4f*)dst)[t] = ((const v4f*)src)[t];
}

__global__ void zero_kernel(float* __restrict__ p, int n) {
  int t = blockIdx.x * blockDim.x + threadIdx.x;
  if (t < n) p[t] = 0.f;
}

// ---------------- fixed-point iterator: f(x) = x - tau*(depthwise3x3(x) - b) ----------------

__global__ void conv_iter_kernel(const float* __restrict__ x, const float* __restrict__ bf,
                                 const float* __restrict__ w, float* __restrict__ out,
                                 float* __restrict__ out2) {
  int idx = blockIdx.x * blockDim.x + threadIdx.x;
  if (idx >= (int)BD_) return;
  int xw = idx % W_;
  int yh = (idx / W_) % H_;
  int c  = (idx / (W_ * H_)) % C_;
  const float* plane = x + (idx - yh * W_ - xw);   // start of this (b,c) 384x384 plane
  float acc = 0.f;
#pragma unroll
  for (int kh = 0; kh < 3; ++kh) {
    int yy = yh + kh - 1;
    if (yy < 0 || yy >= H_) continue;
#pragma unroll
    for (int kw = 0; kw < 3; ++kw) {
      int xx = xw + kw - 1;
      if (xx < 0 || xx >= W_) continue;
      acc += w[c * 9 + kh * 3 + kw] * plane[yy * W_ + xx];
    }
  }
  float v = x[idx] - TAU_ * (acc - bf[idx]);
  out[idx] = v;
  if (out2) out2[idx] = v;
}

// ---------------- Gram: GG[b][i][j] = sum_d G[b,i,d]*G[b,j,d], G = F - X ----------------
// V_WMMA_F32_16X16X4_F32: A 16x4 (lane = M + 16*(K>=2), vgpr = K%2),
// B 4x16 row-striped (lane = N + 16*(K>=2), vgpr = K%2) -> same register image for G*G^T.
// C/D 16x16 f32: vgpr r, lane l -> M = r + 8*(l>=16), N = l%16.
// Each wave owns 4 consecutive K-chunks (64B per lane row) per step: 8 b64 loads in
// one clause at immediate offsets, 4 WMMAs into 4 independent accumulators.

#define GRAM_BLOCKS_PER_BATCH 32   // 256 waves/batch; 110592 chunks / (256*4) = 108 exact

__global__ void gram_kernel(const float* __restrict__ Xh, const float* __restrict__ Fh,
                            float* __restrict__ GG, int n) {
  int b    = blockIdx.x / GRAM_BLOCKS_PER_BATCH;
  int blk  = blockIdx.x % GRAM_BLOCKS_PER_BATCH;
  int lane = threadIdx.x & 31;
  int wave = threadIdx.x >> 5;
  const int wavesPerBatch = GRAM_BLOCKS_PER_BATCH * 8;   // 8 waves per 256-thread block
  int waveId = blk * 8 + wave;

  int row   = lane & 15;   // history slot (M for A, N for B)
  int khalf = lane >> 4;   // which half of the K=4 chunk
  bool active = row < n;
  float msk = active ? 1.f : 0.f;
  size_t rowOff = (size_t)(active ? row : 0) * BD_ + (size_t)b * (size_t)D_;

  // lane base: 4 consecutive chunks (16 floats) per wave step
  const float* fp = Fh + rowOff + (size_t)waveId * 16 + khalf * 2;
  const float* xp = Xh + rowOff + (size_t)waveId * 16 + khalf * 2;
  const size_t stride = (size_t)wavesPerBatch * 16;      // floats advanced per step
  const int iters = (D_ / 4) / (wavesPerBatch * 4);      // = 108, exact

  v8f acc0 = {}, acc1 = {}, acc2 = {}, acc3 = {};
  for (int i = 0; i < iters; ++i) {
    v2f f0 = *(const v2f*)(fp + 0);
    v2f f1 = *(const v2f*)(fp + 4);
    v2f f2 = *(const v2f*)(fp + 8);
    v2f f3 = *(const v2f*)(fp + 12);
    v2f x0 = *(const v2f*)(xp + 0);
    v2f x1 = *(const v2f*)(xp + 4);
    v2f x2 = *(const v2f*)(xp + 8);
    v2f x3 = *(const v2f*)(xp + 12);
    v2f a0 = (f0 - x0) * msk;
    v2f a1 = (f1 - x1) * msk;
    v2f a2 = (f2 - x2) * msk;
    v2f a3 = (f3 - x3) * msk;
    acc0 = __builtin_amdgcn_wmma_f32_16x16x4_f32(false, a0, false, a0, (short)0, acc0, false, false);
    acc1 = __builtin_amdgcn_wmma_f32_16x16x4_f32(false, a1, false, a1, (short)0, acc1, false, false);
    acc2 = __builtin_amdgcn_wmma_f32_16x16x4_f32(false, a2, false, a2, (short)0, acc2, false, false);
    acc3 = __builtin_amdgcn_wmma_f32_16x16x4_f32(false, a3, false, a3, (short)0, acc3, false, false);
    fp += stride;
    xp += stride;
  }
  v8f acc = (acc0 + acc1) + (acc2 + acc3);

  int N = lane & 15;
#pragma unroll
  for (int r = 0; r < 8; ++r) {
    int M = r + 8 * khalf;
    atomicAdd(&GG[(b * 16 + M) * 16 + N], acc[r]);
  }
}

// ---------------- tiny KKT solve per batch: H alpha' = e0, alpha = alpha'[1:] ----------------

__global__ void solve_kernel(const float* __restrict__ GG, float* __restrict__ alpha, int n) {
  int b = threadIdx.x;
  if (b >= B_) return;
  const int m = n + 1;            // <= 6
  float A[6][7];
  for (int i = 0; i < m; ++i) {
    for (int j = 0; j < m; ++j) {
      float v;
      if (i == 0 && j == 0)       v = 0.f;
      else if (i == 0 || j == 0)  v = 1.f;
      else v = GG[(b * 16 + (i - 1)) * 16 + (j - 1)] + ((i == j) ? RIDGE_ : 0.f);
      A[i][j] = v;
    }
    A[i][m] = (i == 0) ? 1.f : 0.f;   // y = e0
  }
  // Gauss-Jordan with partial pivoting (H[0][0]==0 forces a pivot swap)
  for (int col = 0; col < m; ++col) {
    int piv = col;
    float best = fabsf(A[col][col]);
    for (int r = col + 1; r < m; ++r) {
      float v = fabsf(A[r][col]);
      if (v > best) { best = v; piv = r; }
    }
    if (piv != col)
      for (int j = 0; j <= m; ++j) { float t = A[col][j]; A[col][j] = A[piv][j]; A[piv][j] = t; }
    float inv = 1.f / A[col][col];
    for (int j = 0; j <= m; ++j) A[col][j] *= inv;
    for (int r = 0; r < m; ++r) {
      if (r == col) continue;
      float f = A[r][col];
      for (int j = 0; j <= m; ++j) A[r][j] -= f * A[col][j];
    }
  }
  for (int i = 0; i < HIST_; ++i)
    alpha[b * 8 + i] = (i < n) ? A[i + 1][m] : 0.f;
}

// ---------------- xk = sum_i alpha[b][i] * F[b][i][:]  (BETA = 1) ----------------

__global__ void combine_kernel(const float* __restrict__ Fh, const float* __restrict__ alpha,
                               float* __restrict__ xslot, int n) {
  size_t t = (size_t)blockIdx.x * blockDim.x + threadIdx.x;
  if (t >= BD_ / 4) return;
  size_t base = t * 4;
  int b = (int)(base / (size_t)D_);
  v4f s = {};
  for (int i = 0; i < n; ++i) {
    float a = alpha[b * 8 + i];
    v4f f = *(const v4f*)(Fh + (size_t)i * BD_ + base);
    s += f * a;
  }
  *(v4f*)(xslot + base) = s;
}

// ---------------- driver ----------------

extern "C" void kernel_launch(void* const* d_in, const int* in_sizes, int n_in,
                              void* d_out, int out_size, void* d_ws, size_t ws_size,
                              hipStream_t stream) {
  (void)in_sizes; (void)n_in; (void)out_size; (void)ws_size;
  const float* x0 = (const float*)d_in[0];
  const float* bf = (const float*)d_in[1];
  const float* w  = (const float*)d_in[2];   // 27 weights, (3,1,3,3)

  float* Xh    = (float*)d_ws;                       // [HIST][B*D] slot-major
  float* Fh    = Xh + (size_t)HIST_ * BD_;           // [HIST][B*D]
  float* GG    = Fh + (size_t)HIST_ * BD_;           // [B][16][16]
  float* alpha = GG + B_ * 16 * 16;                  // [B][8]

  const int TPB = 256;
  const size_t n4 = BD_ / 4;
  const int copyBlocks = (int)((n4 + TPB - 1) / TPB);
  const int convBlocks = (int)((BD_ + TPB - 1) / TPB);

  // init: X[0]=x0; F[0]=f(x0); X[1]=F[0]; F[1]=f(X[1])
  copy4_kernel<<<copyBlocks, TPB, 0, stream>>>(x0, Xh, n4);
  conv_iter_kernel<<<convBlocks, TPB, 0, stream>>>(x0, bf, w, Fh, Xh + BD_);
  conv_iter_kernel<<<convBlocks, TPB, 0, stream>>>(Xh + BD_, bf, w, Fh + BD_, nullptr);

  for (int it = 2; it < MAXIT_; ++it) {
    int n = it < HIST_ ? it : HIST_;
    int slot = it % HIST_;
    zero_kernel<<<(B_ * 256 + TPB - 1) / TPB, TPB, 0, stream>>>(GG, B_ * 256);
    gram_kernel<<<B_ * GRAM_BLOCKS_PER_BATCH, TPB, 0, stream>>>(Xh, Fh, GG, n);
    solve_kernel<<<1, 32, 0, stream>>>(GG, alpha, n);
    combine_kernel<<<copyBlocks, TPB, 0, stream>>>(Fh, alpha, Xh + (size_t)slot * BD_, n);
    conv_iter_kernel<<<convBlocks, TPB, 0, stream>>>(Xh + (size_t)slot * BD_, bf, w,
                                                     Fh + (size_t)slot * BD_, nullptr);
  }

  // output = F[(MAXIT-1) % HIST] = F[1]
  copy4_kernel<<<copyBlocks, TPB, 0, stream>>>(Fh + BD_, (float*)d_out, n4);
}